// EmbeddingWithConv2d_19018115187037
// MI455X (gfx1250) — compile-verified
//
#include <hip/hip_runtime.h>

// Embedding-as-1x1-conv == column gather from W [EMBED, VOCAB] + bias.
// out[t, e] = W[e * VOCAB + idx[t]] + b[e],  t in [0, 8192), e in [0, 512)

#define VOCAB 32000
#define EMBED 512

// 64 threads per token; each thread produces 8 consecutive embed elements
// (two float4 stores). 8 independent strided gather loads per thread give
// enough memory-level parallelism to hide HBM latency; all W-column strides
// (k * VOCAB * 4 bytes, k=0..7, max 896000) fit in the 24-bit signed
// immediate offset of global_load_b32, so only one VGPR address per thread.
__global__ __launch_bounds__(256) void embed_gather_kernel(
    const int* __restrict__ idx,
    const float* __restrict__ W,
    const float* __restrict__ bias,
    float* __restrict__ out,
    int n_tokens)
{
    const int gid   = blockIdx.x * blockDim.x + threadIdx.x;
    const int token = gid >> 6;          // 64 threads per token
    const int e0    = (gid & 63) << 3;   // 8 embed elements per thread

    if (token >= n_tokens) return;

    // Prefetch the index cacheline a bit ahead (gfx1250 global_prefetch_b8).
    if ((gid & 63) == 0 && (token + 64) < n_tokens) {
        __builtin_prefetch(idx + token + 64, 0, 0);
    }

    const int v = idx[token];            // wave-uniform (64-thread slices)

    // Base of this thread's strided column segment: W[e0 * VOCAB + v]
    const float* __restrict__ col = W + (size_t)e0 * VOCAB + (size_t)v;

    float r[8];
#pragma unroll
    for (int k = 0; k < 8; ++k) {
        r[k] = col[(size_t)k * VOCAB];   // immediate-offset gathers
    }

    const float4 b0 = *(const float4*)(bias + e0);
    const float4 b1 = *(const float4*)(bias + e0 + 4);

    float4 o0, o1;
    o0.x = r[0] + b0.x;  o0.y = r[1] + b0.y;
    o0.z = r[2] + b0.z;  o0.w = r[3] + b0.w;
    o1.x = r[4] + b1.x;  o1.y = r[5] + b1.y;
    o1.z = r[6] + b1.z;  o1.w = r[7] + b1.w;

    float* __restrict__ dst = out + (size_t)token * EMBED + e0;
    *(float4*)(dst)     = o0;            // global_store_b128, fully coalesced
    *(float4*)(dst + 4) = o1;
}

extern "C" void kernel_launch(void* const* d_in, const int* in_sizes, int n_in,
                              void* d_out, int out_size, void* d_ws, size_t ws_size,
                              hipStream_t stream) {
    const int*   idx  = (const int*)d_in[0];     // [4*2048] token ids
    const float* W    = (const float*)d_in[1];   // [EMBED, VOCAB] row-major
    const float* bias = (const float*)d_in[2];   // [EMBED]
    float*       out  = (float*)d_out;           // [tokens, EMBED]

    const int n_tokens = in_sizes[0];            // 8192
    const int threads  = n_tokens * 64;          // 64 threads per token
    const int block    = 256;                    // 8 waves (wave32) per block
    const int grid     = (threads + block - 1) / block;

    embed_gather_kernel<<<grid, block, 0, stream>>>(idx, W, bias, out, n_tokens);
}